// GCNEncoder_89627377533231
// MI455X (gfx1250) — compile-verified
//
#include <hip/hip_runtime.h>

#define N_NODES 50000
#define N_EDGES 500000
#define IN_C 512
#define HID_C 256
#define OUT_C 128

typedef __attribute__((ext_vector_type(2))) float v2f;
typedef __attribute__((ext_vector_type(8))) float v8f;

// ---------------------------------------------------------------------------
// Degree / normalization precompute
// ---------------------------------------------------------------------------
__global__ void init_deg_kernel(float* __restrict__ deg, int n) {
  int i = blockIdx.x * blockDim.x + threadIdx.x;
  if (i < n) deg[i] = 1.0f;  // self-loop contributes 1
}

__global__ void deg_atomic_kernel(const int* __restrict__ dst, float* __restrict__ deg, int e) {
  int i = blockIdx.x * blockDim.x + threadIdx.x;
  if (i < e) unsafeAtomicAdd(&deg[dst[i]], 1.0f);
}

__global__ void norm_kernel(const float* __restrict__ deg, float* __restrict__ dis,
                            float* __restrict__ inv, int n) {
  int i = blockIdx.x * blockDim.x + threadIdx.x;
  if (i < n) {
    float d = deg[i];
    dis[i] = rsqrtf(d);   // D^{-1/2}
    inv[i] = 1.0f / d;    // self-loop coefficient
  }
}

__global__ void coef_kernel(const int* __restrict__ src, const int* __restrict__ dst,
                            const float* __restrict__ dis, float* __restrict__ coef, int e) {
  int i = blockIdx.x * blockDim.x + threadIdx.x;
  if (i < e) coef[i] = dis[src[i]] * dis[dst[i]];
}

// ---------------------------------------------------------------------------
// Weight transpose: B[K][N] -> Bt[N][K]  (tiny, one-shot)
// Makes the WMMA B fragment a contiguous float2 load, same shape as A.
// ---------------------------------------------------------------------------
__global__ void transpose_kernel(const float* __restrict__ B, float* __restrict__ Bt,
                                 int K, int N) {
  int i = blockIdx.x * blockDim.x + threadIdx.x;
  if (i < K * N) {
    int kk = i / N;
    int n = i - kk * N;
    Bt[(size_t)n * K + kk] = B[i];
  }
}

// ---------------------------------------------------------------------------
// fp32 WMMA GEMM: C[M,N] = A[M,K] * Bt[N,K]^T, row-major.
// One wave computes a (MT*16) x (NT*16) register tile with V_WMMA_F32_16X16X4_F32.
// Per k-step: MT float2 A-frag loads + NT float2 B-frag loads feed MT*NT WMMAs.
// Fragment layouts (wave32):
//   A (16x4): lanes 0-15 -> row M=l, K pair (k,k+1); lanes 16-31 -> K pair (k+2,k+3)
//   B (4x16): same striping on Bt rows (n major) -> contiguous float2 at Bt[n][ka]
//   C (16x16): VGPR r -> row r (lanes 0-15) / row r+8 (lanes 16-31), col = l
// Out-of-range M handled by clamped loads + wave-uniform store guards so EXEC
// stays all-ones through every WMMA (ISA requirement).
// ---------------------------------------------------------------------------
template <int MT, int NT>
__global__ __launch_bounds__(128) void wmma_gemm_f32(const float* __restrict__ A,
                                                     const float* __restrict__ Bt,
                                                     float* __restrict__ C,
                                                     int M, int N, int K) {
  const int wave = (blockIdx.x * blockDim.x + threadIdx.x) >> 5;
  const int lane = threadIdx.x & 31;
  const int ngroups = N / (NT * 16);
  const int sm = wave / ngroups;          // super row-tile (MT*16 rows)
  const int tg = wave % ngroups;          // column group
  const int m0 = sm * MT * 16;
  if (m0 >= M) return;                    // whole-wave exit

  const int half = lane >> 4;             // 0: K pair (k,k+1); 1: (k+2,k+3)
  const int l = lane & 15;

  const float* a_ptr[MT];
#pragma unroll
  for (int i = 0; i < MT; ++i) {
    int row = m0 + i * 16 + l;
    if (row >= M) row = M - 1;            // clamp: load valid memory, discard result
    a_ptr[i] = A + (size_t)row * K;
  }
  const float* b_ptr[NT];
#pragma unroll
  for (int t = 0; t < NT; ++t) {
    int col = tg * (NT * 16) + t * 16 + l;
    b_ptr[t] = Bt + (size_t)col * K;
  }

  v8f acc[MT][NT];
#pragma unroll
  for (int i = 0; i < MT; ++i)
#pragma unroll
    for (int t = 0; t < NT; ++t) acc[i][t] = (v8f){0, 0, 0, 0, 0, 0, 0, 0};

  for (int k = 0; k < K; k += 4) {
    const int ka = k + 2 * half;
    v2f a[MT], b[NT];
#pragma unroll
    for (int i = 0; i < MT; ++i) a[i] = *(const v2f*)(a_ptr[i] + ka);
#pragma unroll
    for (int t = 0; t < NT; ++t) b[t] = *(const v2f*)(b_ptr[t] + ka);
#pragma unroll
    for (int i = 0; i < MT; ++i)
#pragma unroll
      for (int t = 0; t < NT; ++t)
        acc[i][t] = __builtin_amdgcn_wmma_f32_16x16x4_f32(
            false, a[i], false, b[t], (short)0, acc[i][t], false, false);
  }

#pragma unroll
  for (int i = 0; i < MT; ++i) {
    if (m0 + i * 16 + 16 <= M) {          // wave-uniform guard
      float* c_ptr = C + (size_t)(m0 + i * 16) * N + (size_t)tg * (NT * 16) + l;
#pragma unroll
      for (int t = 0; t < NT; ++t)
#pragma unroll
        for (int r = 0; r < 8; ++r)
          c_ptr[(size_t)(r + 8 * half) * N + t * 16] = acc[i][t][r];
    }
  }
}

// ---------------------------------------------------------------------------
// agg = z * self_norm[node] + bias[ch]     (fully initializes the buffer)
// ---------------------------------------------------------------------------
__global__ void init_agg_kernel(float* __restrict__ agg, const float* __restrict__ z,
                                const float* __restrict__ inv, const float* __restrict__ bias,
                                int n, int c) {
  int i = blockIdx.x * blockDim.x + threadIdx.x;
  if (i < n * c) {
    int node = i / c;
    int ch = i - node * c;
    agg[i] = z[i] * inv[node] + bias[ch];
  }
}

// ---------------------------------------------------------------------------
// agg[dst] += z[src] * coef   -- edge-parallel, 4 channels per thread,
// float4 gather + 4 hardware FP32 global atomics (L2-resident working set)
// ---------------------------------------------------------------------------
__global__ void edge_agg_kernel(float* __restrict__ agg, const float* __restrict__ z,
                                const int* __restrict__ src, const int* __restrict__ dst,
                                const float* __restrict__ coef, int e, int c) {
  int i = blockIdx.x * blockDim.x + threadIdx.x;
  const int groups = c >> 2;
  if (i >= e * groups) return;
  const int eg = i / groups;
  const int g = i - eg * groups;
  const int s = src[eg];
  const int d = dst[eg];
  const float w = coef[eg];
  const float4 v = *(const float4*)(z + (size_t)s * c + g * 4);
  float* out = agg + (size_t)d * c + g * 4;
  unsafeAtomicAdd(out + 0, v.x * w);
  unsafeAtomicAdd(out + 1, v.y * w);
  unsafeAtomicAdd(out + 2, v.z * w);
  unsafeAtomicAdd(out + 3, v.w * w);
}

__global__ void relu_kernel(float* __restrict__ h, int total) {
  int i = blockIdx.x * blockDim.x + threadIdx.x;
  if (i < total) h[i] = fmaxf(h[i], 0.0f);
}

// ---------------------------------------------------------------------------
// Host-side orchestration
// ---------------------------------------------------------------------------
static inline int cdiv_l(long a, long b) { return (int)((a + b - 1) / b); }

extern "C" void kernel_launch(void* const* d_in, const int* in_sizes, int n_in,
                              void* d_out, int out_size, void* d_ws, size_t ws_size,
                              hipStream_t stream) {
  const float* x   = (const float*)d_in[0];
  const int*   ei  = (const int*)d_in[1];
  const int*   src = ei;                 // edge_index[0]
  const int*   dst = ei + N_EDGES;       // edge_index[1]
  const float* w1  = (const float*)d_in[2];
  const float* b1  = (const float*)d_in[3];
  const float* wmu = (const float*)d_in[4];
  const float* bmu = (const float*)d_in[5];
  const float* wls = (const float*)d_in[6];
  const float* bls = (const float*)d_in[7];

  float* out_mu = (float*)d_out;
  float* out_ls = out_mu + (size_t)N_NODES * OUT_C;

  float* ws = (float*)d_ws;
  size_t off = 0;
  float* deg  = ws + off; off += N_NODES;
  float* dis  = ws + off; off += N_NODES;
  float* inv  = ws + off; off += N_NODES;
  float* coef = ws + off; off += N_EDGES;
  float* w1t  = ws + off; off += (size_t)IN_C * HID_C;     // w1^T   [HID][IN]
  float* wmut = ws + off; off += (size_t)HID_C * OUT_C;    // w_mu^T [OUT][HID]
  float* wlst = ws + off; off += (size_t)HID_C * OUT_C;    // w_ls^T [OUT][HID]
  float* hpre = ws + off; off += (size_t)N_NODES * HID_C;  // x @ w1
  float* h    = ws + off; off += (size_t)N_NODES * HID_C;  // relu(conv1)
  float* zmu  = ws + off; off += (size_t)N_NODES * OUT_C;  // h @ w_mu
  float* zls  = ws + off; off += (size_t)N_NODES * OUT_C;  // h @ w_ls

  // --- normalization precompute ---
  init_deg_kernel<<<cdiv_l(N_NODES, 256), 256, 0, stream>>>(deg, N_NODES);
  deg_atomic_kernel<<<cdiv_l(N_EDGES, 256), 256, 0, stream>>>(dst, deg, N_EDGES);
  norm_kernel<<<cdiv_l(N_NODES, 256), 256, 0, stream>>>(deg, dis, inv, N_NODES);
  coef_kernel<<<cdiv_l(N_EDGES, 256), 256, 0, stream>>>(src, dst, dis, coef, N_EDGES);

  // --- one-shot weight transposes (tiny) ---
  transpose_kernel<<<cdiv_l((long)IN_C * HID_C, 256), 256, 0, stream>>>(w1, w1t, IN_C, HID_C);
  transpose_kernel<<<cdiv_l((long)HID_C * OUT_C, 256), 256, 0, stream>>>(wmu, wmut, HID_C, OUT_C);
  transpose_kernel<<<cdiv_l((long)HID_C * OUT_C, 256), 256, 0, stream>>>(wls, wlst, HID_C, OUT_C);

  constexpr int MT = 4, NT = 4;  // 64x64 output tile per wave
  const long m_super = cdiv_l(N_NODES, MT * 16);

  // --- conv1: h = relu(Ahat * (x @ w1) + b1) ---
  {
    long waves = m_super * (HID_C / (NT * 16));
    wmma_gemm_f32<MT, NT><<<cdiv_l(waves * 32, 128), 128, 0, stream>>>(
        x, w1t, hpre, N_NODES, HID_C, IN_C);
  }
  init_agg_kernel<<<cdiv_l((long)N_NODES * HID_C, 256), 256, 0, stream>>>(h, hpre, inv, b1,
                                                                          N_NODES, HID_C);
  edge_agg_kernel<<<cdiv_l((long)N_EDGES * (HID_C / 4), 256), 256, 0, stream>>>(
      h, hpre, src, dst, coef, N_EDGES, HID_C);
  relu_kernel<<<cdiv_l((long)N_NODES * HID_C, 256), 256, 0, stream>>>(h, N_NODES * HID_C);

  // --- mu = Ahat * (h @ w_mu) + b_mu   (accumulated directly into d_out) ---
  {
    long waves = m_super * (OUT_C / (NT * 16));
    wmma_gemm_f32<MT, NT><<<cdiv_l(waves * 32, 128), 128, 0, stream>>>(
        h, wmut, zmu, N_NODES, OUT_C, HID_C);
  }
  init_agg_kernel<<<cdiv_l((long)N_NODES * OUT_C, 256), 256, 0, stream>>>(out_mu, zmu, inv, bmu,
                                                                          N_NODES, OUT_C);
  edge_agg_kernel<<<cdiv_l((long)N_EDGES * (OUT_C / 4), 256), 256, 0, stream>>>(
      out_mu, zmu, src, dst, coef, N_EDGES, OUT_C);

  // --- logstd = Ahat * (h @ w_ls) + b_ls ---
  {
    long waves = m_super * (OUT_C / (NT * 16));
    wmma_gemm_f32<MT, NT><<<cdiv_l(waves * 32, 128), 128, 0, stream>>>(
        h, wlst, zls, N_NODES, OUT_C, HID_C);
  }
  init_agg_kernel<<<cdiv_l((long)N_NODES * OUT_C, 256), 256, 0, stream>>>(out_ls, zls, inv, bls,
                                                                          N_NODES, OUT_C);
  edge_agg_kernel<<<cdiv_l((long)N_EDGES * (OUT_C / 4), 256), 256, 0, stream>>>(
      out_ls, zls, src, dst, coef, N_EDGES, OUT_C);
}